// MotionCorrectBatch_10264971837757
// MI455X (gfx1250) — compile-verified
//
#include <hip/hip_runtime.h>
#include <math.h>

#define B_    32
#define H_    512
#define W_    512
#define HW_   (H_ * W_)        // 262144
#define NPIX_ (B_ * HW_)       // 8388608
#define MS_   10
#define Y_    21               // 2*MS+1
#define KH_   (H_ - 2 * MS_)   // 492
#define EPS_  1e-8f

typedef __attribute__((ext_vector_type(2))) float v2f;
typedef __attribute__((ext_vector_type(8))) float v8f;

// ---------------------------------------------------------------------------
// 512-point complex DFT along one axis, implemented as GEMM with the DFT
// matrix using V_WMMA_F32_16X16X4_F32. One wave computes a 16x16 complex
// output tile: Out(16 freqs x 16 transforms), K=512 accumulated in 128
// chunks of 4. Twiddles computed on the fly with exact mod-512 phase.
// Element (t, j) of the input lives at  (t>>9)*HW + (t&511)*strideT + j*strideJ.
// ---------------------------------------------------------------------------
__global__ __launch_bounds__(32) void dft512_wmma_kernel(
    const float* __restrict__ ir, const float* __restrict__ ii,
    float* __restrict__ out_r, float* __restrict__ out_i,
    int strideT, int strideJ, float sign, float scale)
{
  const int lane = threadIdx.x;      // 0..31
  const int half = lane >> 4;        // 0 or 1
  const int l    = lane & 15;
  const int k0 = blockIdx.x * 16;    // frequency tile
  const int t0 = blockIdx.y * 16;    // transform tile

  const int t     = t0 + l;
  const int tbase = (t >> 9) * HW_ + (t & 511) * strideT;

  v8f accRR = {}, accII = {}, accRI = {}, accIR = {};
  const float w0 = sign * 6.283185307179586f / 512.0f;
  const int   m  = k0 + l;           // A-matrix row (freq) handled by this lane

  for (int kc = 0; kc < 128; ++kc) {
    const int j0 = kc * 4 + 2 * half;   // ISA A/B layout: lane-half selects K pair
    const int j1 = j0 + 1;

    // A tile: DFT weights W[m][j] = exp(i*sign*2*pi*m*j/512), phase mod 512
    float c0, s0, c1, s1;
    __sincosf(w0 * (float)((m * j0) & 511), &s0, &c0);
    __sincosf(w0 * (float)((m * j1) & 511), &s1, &c1);
    v2f ar; ar.x = c0; ar.y = c1;
    v2f ai; ai.x = s0; ai.y = s1;

    // B tile: data X[j][n], n = transform index = l
    const int o0 = tbase + j0 * strideJ;
    const int o1 = tbase + j1 * strideJ;
    if (kc + 16 < 128 && (kc & 15) == 0) {
      __builtin_prefetch(&ir[tbase + (j0 + 64) * strideJ], 0, 1);
      __builtin_prefetch(&ii[tbase + (j0 + 64) * strideJ], 0, 1);
    }
    v2f br; br.x = ir[o0]; br.y = ir[o1];
    v2f bi; bi.x = ii[o0]; bi.y = ii[o1];

    // Complex GEMM: Or = Wr*Xr - Wi*Xi ; Oi = Wr*Xi + Wi*Xr
    accRR = __builtin_amdgcn_wmma_f32_16x16x4_f32(false, ar, false, br, (short)0, accRR, false, false);
    accII = __builtin_amdgcn_wmma_f32_16x16x4_f32(false, ai, false, bi, (short)0, accII, false, false);
    accRI = __builtin_amdgcn_wmma_f32_16x16x4_f32(false, ar, false, bi, (short)0, accRI, false, false);
    accIR = __builtin_amdgcn_wmma_f32_16x16x4_f32(false, ai, false, br, (short)0, accIR, false, false);
  }

  // D layout: VGPR v holds (M=v, N=l) for lanes 0-15 and (M=v+8, N=l) for 16-31
#pragma unroll
  for (int v = 0; v < 8; ++v) {
    const int mm = v + 8 * half;
    const int o  = tbase + (k0 + mm) * strideJ;
    out_r[o] = scale * (accRR[v] - accII[v]);
    out_i[o] = scale * (accRI[v] + accIR[v]);
  }
}

// ---------------------------------------------------------------------------
// Per-image mean (B blocks)
// ---------------------------------------------------------------------------
__global__ void image_mean_kernel(const float* __restrict__ f, float* __restrict__ means)
{
  __shared__ float red[256];
  const int b = blockIdx.x;
  float s = 0.f;
  for (int i = threadIdx.x; i < HW_; i += 256) s += f[b * HW_ + i];
  red[threadIdx.x] = s;
  __syncthreads();
  for (int st = 128; st > 0; st >>= 1) {
    if (threadIdx.x < st) red[threadIdx.x] += red[threadIdx.x + st];
    __syncthreads();
  }
  if (threadIdx.x == 0) means[b] = red[0] / (float)HW_;
}

// Template mean + zero-mean variance (+EPS), single block
__global__ void template_stats_kernel(const float* __restrict__ t, float* __restrict__ tstats)
{
  __shared__ float red[256];
  __shared__ float meansh;
  float s = 0.f;
  for (int i = threadIdx.x; i < HW_; i += 256) s += t[i];
  red[threadIdx.x] = s;
  __syncthreads();
  for (int st = 128; st > 0; st >>= 1) {
    if (threadIdx.x < st) red[threadIdx.x] += red[threadIdx.x + st];
    __syncthreads();
  }
  if (threadIdx.x == 0) meansh = red[0] / (float)HW_;
  __syncthreads();
  const float mean = meansh;
  float s2 = 0.f;
  for (int i = threadIdx.x; i < HW_; i += 256) { float v = t[i] - mean; s2 += v * v; }
  red[threadIdx.x] = s2;
  __syncthreads();
  for (int st = 128; st > 0; st >>= 1) {
    if (threadIdx.x < st) red[threadIdx.x] += red[threadIdx.x + st];
    __syncthreads();
  }
  if (threadIdx.x == 0) { tstats[0] = mean; tstats[1] = red[0] + EPS_; }
}

__global__ void pack_images_kernel(const float* __restrict__ f, const float* __restrict__ means,
                                   float* __restrict__ ar, float* __restrict__ ai)
{
  const int i = blockIdx.x * 256 + threadIdx.x;
  if (i >= NPIX_) return;
  ar[i] = f[i] - means[i / HW_];
  ai[i] = 0.f;
}

__global__ void pack_template_kernel(const float* __restrict__ t, const float* __restrict__ tstats,
                                     float* __restrict__ tr, float* __restrict__ ti)
{
  const int i = blockIdx.x * 256 + threadIdx.x;
  if (i >= HW_) return;
  tr[i] = t[i] - tstats[0];
  ti[i] = 0.f;
}

// ---------------------------------------------------------------------------
// Fused: forward length-32 DFT along batch axis, multiply by conj(Tfreq),
// inverse length-32 DFT. All register resident per pixel (the two batch
// transforms bracket only a pointwise multiply).
// ---------------------------------------------------------------------------
__global__ void batch_fft_mul_ifft_kernel(const float* __restrict__ ir, const float* __restrict__ ii,
                                          float* __restrict__ out_r, float* __restrict__ out_i,
                                          const float* __restrict__ tr, const float* __restrict__ ti)
{
  __shared__ float ct[32], st[32];
  if (threadIdx.x < 32) {
    float th = 6.283185307179586f / 32.f * (float)threadIdx.x;
    ct[threadIdx.x] = __cosf(th);
    st[threadIdx.x] = __sinf(th);
  }
  __syncthreads();
  const int p = blockIdx.x * blockDim.x + threadIdx.x;
  if (p >= HW_) return;

  float xr[32], xi[32], yr[32], yi[32];
  for (int b = 0; b < 32; ++b) { xr[b] = ir[b * HW_ + p]; xi[b] = ii[b * HW_ + p]; }

  // forward: exp(-i*2pi*k*n/32)
  for (int k = 0; k < 32; ++k) {
    float sr = 0.f, si = 0.f;
    for (int n = 0; n < 32; ++n) {
      const int idx = (k * n) & 31;
      const float c = ct[idx], s = -st[idx];
      sr += c * xr[n] - s * xi[n];
      si += c * xi[n] + s * xr[n];
    }
    yr[k] = sr; yi[k] = si;
  }
  // multiply by conj(T) and fold in 1/32 of inverse normalization
  const float trv = tr[p], tiv = ti[p];
  for (int k = 0; k < 32; ++k) {
    const float a = yr[k], b = yi[k];
    yr[k] = (a * trv + b * tiv) * (1.f / 32.f);
    yi[k] = (b * trv - a * tiv) * (1.f / 32.f);
  }
  // inverse: exp(+i*2pi*b*n/32)
  for (int b = 0; b < 32; ++b) {
    float sr = 0.f, si = 0.f;
    for (int n = 0; n < 32; ++n) {
      const int idx = (b * n) & 31;
      const float c = ct[idx], s = st[idx];
      sr += c * yr[n] - s * yi[n];
      si += c * yi[n] + s * yr[n];
    }
    out_r[b * HW_ + p] = sr;
    out_i[b * HW_ + p] = si;
  }
}

// ---------------------------------------------------------------------------
// Separable 492x492 window sums (only 21x21 positions needed)
// ---------------------------------------------------------------------------
__global__ void hsum_kernel(const float* __restrict__ f,
                            float* __restrict__ hs, float* __restrict__ hq)
{
  const int idx = blockIdx.x * 256 + threadIdx.x;  // (b*512+h)*21 + j
  if (idx >= B_ * H_ * Y_) return;
  const int j  = idx % Y_;
  const int bh = idx / Y_;
  const float* row = f + bh * W_;
  float s = 0.f, q = 0.f;
  for (int w = j; w < j + KH_; ++w) { const float v = row[w]; s += v; q += v * v; }
  hs[idx] = s;
  hq[idx] = q;
}

__global__ void denom_kernel(const float* __restrict__ hs, const float* __restrict__ hq,
                             const float* __restrict__ tstats, float* __restrict__ den)
{
  const int idx = blockIdx.x * 256 + threadIdx.x;  // b*441 + i*21 + j
  if (idx >= B_ * Y_ * Y_) return;
  const int b = idx / (Y_ * Y_);
  const int r = idx % (Y_ * Y_);
  const int i = r / Y_, j = r % Y_;
  float s = 0.f, q = 0.f;
  for (int h = i; h < i + KH_; ++h) {
    const int o = (b * H_ + h) * Y_ + j;
    s += hs[o];
    q += hq[o];
  }
  const float inv = 1.f / ((float)KH_ * (float)KH_);
  const float ls = s * inv, lq = q * inv;
  float var = lq - ls * ls / (float)HW_ + EPS_;
  if (var < 0.f) var = 0.f;
  den[idx] = sqrtf(tstats[1] * var);
}

// ---------------------------------------------------------------------------
// NCC window, argmax (first-max tie-break), log-parabola subpixel shift
// ---------------------------------------------------------------------------
__global__ __launch_bounds__(448) void ncc_shift_kernel(
    const float* __restrict__ cr, const float* __restrict__ ci,
    const float* __restrict__ den, float* __restrict__ shifts)
{
  __shared__ float ncc_s[Y_ * Y_];
  const int b   = blockIdx.x;
  const int tid = threadIdx.x;
  if (tid < Y_ * Y_) {
    const int i   = tid / Y_, j = tid % Y_;
    const int row = (i - MS_ + H_) & (H_ - 1);   // rolled correlation index
    const int col = (j - MS_ + W_) & (W_ - 1);
    const int o   = b * HW_ + row * W_ + col;
    const float v = sqrtf(cr[o] * cr[o] + ci[o] * ci[o]);
    float n = v / den[b * Y_ * Y_ + tid];
    if (isnan(n)) n = 0.f;
    ncc_s[tid] = n;
  }
  __syncthreads();
  if (tid == 0) {
    int am = 0; float best = ncc_s[0];
    for (int k = 1; k < Y_ * Y_; ++k)
      if (ncc_s[k] > best) { best = ncc_s[k]; am = k; }
    const int sx = am / Y_, sy = am % Y_;
    float shx = -(float)(sx - MS_);
    float shy = -(float)(sy - MS_);
    // JAX semantics: negative index wraps, overflow clamps
    int sxm = sx - 1; if (sxm < 0) sxm += Y_;
    int sxp = sx + 1; if (sxp > Y_ - 1) sxp = Y_ - 1;
    int sym = sy - 1; if (sym < 0) sym += Y_;
    int syp = sy + 1; if (syp > Y_ - 1) syp = Y_ - 1;
    const float lxm = __logf(ncc_s[sxm * Y_ + sy]);
    const float lxp = __logf(ncc_s[sxp * Y_ + sy]);
    const float lym = __logf(ncc_s[sx * Y_ + sym]);
    const float lyp = __logf(ncc_s[sx * Y_ + syp]);
    const float l4  = 4.f * __logf(ncc_s[sx * Y_ + sy]);
    shx -= (lxm - lxp) / (2.f * lxm - l4 + 2.f * lxp);
    shy -= (lym - lyp) / (2.f * lym - l4 + 2.f * lyp);
    shifts[b * 2 + 0] = shx;  // dx (sh_x_n)
    shifts[b * 2 + 1] = shy;  // dy (sh_y_n)
  }
}

// ---------------------------------------------------------------------------
// Bilinear translate with zero padding; transposed (b, x, y) writeout
// ---------------------------------------------------------------------------
__global__ void translate_kernel(const float* __restrict__ f, const float* __restrict__ shifts,
                                 float* __restrict__ out)
{
  const int idx = blockIdx.x * 256 + threadIdx.x;
  if (idx >= NPIX_) return;
  const int b = idx / HW_;
  const int r = idx % HW_;
  const int y = r / W_;
  const int x = r % W_;
  const float dx = shifts[b * 2 + 0];
  const float dy = shifts[b * 2 + 1];
  const float sy = (float)y - dy;
  const float sx = (float)x - dx;
  const float y0f = floorf(sy), x0f = floorf(sx);
  const int y0 = (int)y0f, x0 = (int)x0f;
  const float wy = sy - y0f, wx = sx - x0f;
  const float* img = f + b * HW_;
  auto grab = [&](int yi, int xi) -> float {
    const bool valid = (yi >= 0) && (yi < H_) && (xi >= 0) && (xi < W_);
    const int yc = yi < 0 ? 0 : (yi > H_ - 1 ? H_ - 1 : yi);
    const int xc = xi < 0 ? 0 : (xi > W_ - 1 ? W_ - 1 : xi);
    const float v = img[yc * W_ + xc];
    return valid ? v : 0.f;
  };
  const float val = (1.f - wy) * (1.f - wx) * grab(y0, x0)
                  + (1.f - wy) * wx         * grab(y0, x0 + 1)
                  + wy         * (1.f - wx) * grab(y0 + 1, x0)
                  + wy         * wx         * grab(y0 + 1, x0 + 1);
  out[b * HW_ + x * H_ + y] = val;
}

// ---------------------------------------------------------------------------
extern "C" void kernel_launch(void* const* d_in, const int* in_sizes, int n_in,
                              void* d_out, int out_size, void* d_ws, size_t ws_size,
                              hipStream_t stream)
{
  (void)in_sizes; (void)n_in; (void)out_size; (void)ws_size;
  const float* f    = (const float*)d_in[0];   // fr (1,B,H,W,1) == (B,H,W)
  const float* tmpl = (const float*)d_in[1];   // template (H,W)
  float* out = (float*)d_out;

  float* ws = (float*)d_ws;
  size_t off = 0;
  auto alloc = [&](size_t n) { float* p = ws + off; off += n; return p; };
  float* Are    = alloc(NPIX_);
  float* Aim    = alloc(NPIX_);
  float* Bre    = alloc(NPIX_);
  float* Bim    = alloc(NPIX_);
  float* Tre    = alloc(HW_);
  float* Tim    = alloc(HW_);
  float* T2re   = alloc(HW_);
  float* T2im   = alloc(HW_);
  float* means  = alloc(B_);
  float* tstats = alloc(2);
  float* hs     = alloc((size_t)B_ * H_ * Y_);
  float* hq     = alloc((size_t)B_ * H_ * Y_);
  float* den    = alloc((size_t)B_ * Y_ * Y_);
  float* shifts = alloc(2 * B_);

  image_mean_kernel<<<B_, 256, 0, stream>>>(f, means);
  template_stats_kernel<<<1, 256, 0, stream>>>(tmpl, tstats);
  pack_images_kernel<<<(NPIX_ + 255) / 256, 256, 0, stream>>>(f, means, Are, Aim);
  pack_template_kernel<<<(HW_ + 255) / 256, 256, 0, stream>>>(tmpl, tstats, Tre, Tim);

  // Template 2D DFT (512 transforms): rows then cols
  dft512_wmma_kernel<<<dim3(32, 32), 32, 0, stream>>>(Tre, Tim, T2re, T2im, W_, 1, -1.f, 1.f);
  dft512_wmma_kernel<<<dim3(32, 32), 32, 0, stream>>>(T2re, T2im, Tre, Tim, 1, W_, -1.f, 1.f);

  // Images forward 2D DFT (16384 transforms per pass)
  dft512_wmma_kernel<<<dim3(32, 1024), 32, 0, stream>>>(Are, Aim, Bre, Bim, W_, 1, -1.f, 1.f);
  dft512_wmma_kernel<<<dim3(32, 1024), 32, 0, stream>>>(Bre, Bim, Are, Aim, 1, W_, -1.f, 1.f);

  // Batch-axis fwd DFT * conj(T) * (1/32) -> batch-axis inverse DFT, fused
  batch_fft_mul_ifft_kernel<<<(HW_ + 255) / 256, 256, 0, stream>>>(Are, Aim, Bre, Bim, Tre, Tim);

  // Inverse 2D DFT (cols, rows), each with 1/512 normalization
  dft512_wmma_kernel<<<dim3(32, 1024), 32, 0, stream>>>(Bre, Bim, Are, Aim, 1, W_, +1.f, 1.f / 512.f);
  dft512_wmma_kernel<<<dim3(32, 1024), 32, 0, stream>>>(Are, Aim, Bre, Bim, W_, 1, +1.f, 1.f / 512.f);

  // Local variance denominator (separable window sums)
  hsum_kernel<<<(B_ * H_ * Y_ + 255) / 256, 256, 0, stream>>>(f, hs, hq);
  denom_kernel<<<(B_ * Y_ * Y_ + 255) / 256, 256, 0, stream>>>(hs, hq, tstats, den);

  // NCC window + argmax + subpixel shifts
  ncc_shift_kernel<<<B_, 448, 0, stream>>>(Bre, Bim, den, shifts);

  // Bilinear translate + transposed writeout
  translate_kernel<<<(NPIX_ + 255) / 256, 256, 0, stream>>>(f, shifts, out);
}